// SoftmaxAutoweightedTotalLoss_29231547417214
// MI455X (gfx1250) — compile-verified
//
#include <hip/hip_runtime.h>

#define C_DIM   1024
#define EPSV    1e-5f

typedef __attribute__((ext_vector_type(2))) float v2f;
typedef __attribute__((ext_vector_type(8))) float v8f;

// ---------------- workspace layout (floats/ints, 4B units) ----------------
//  [0 .. 1023]      : int   counts[1024]
//  [1024]           : float min_count
//  [1088 .. 1088+8191] : float partials[B/8]
// --------------------------------------------------------------------------

__global__ void zero_counts_kernel(int* __restrict__ counts) {
    int i = blockIdx.x * blockDim.x + threadIdx.x;
    if (i < C_DIM) counts[i] = 0;
}

__global__ void hist_kernel(const int* __restrict__ target, int* __restrict__ counts, int B) {
    int i = blockIdx.x * blockDim.x + threadIdx.x;
    if (i < B) atomicAdd(&counts[target[i]], 1);
}

__global__ void __launch_bounds__(256)
min_kernel(const int* __restrict__ counts, float* __restrict__ minv) {
    int m = 0x7fffffff;
    for (int i = threadIdx.x; i < C_DIM; i += 256) m = min(m, counts[i]);
    #pragma unroll
    for (int off = 16; off; off >>= 1) m = min(m, __shfl_xor(m, off, 32));
    __shared__ int sm[8];
    if ((threadIdx.x & 31) == 0) sm[threadIdx.x >> 5] = m;
    __syncthreads();
    if (threadIdx.x == 0) {
        int r = sm[0];
        #pragma unroll
        for (int i = 1; i < 8; ++i) r = min(r, sm[i]);
        *minv = (float)r;
    }
}

// One wave32 per row: 8 x float4 coalesced loads keep the whole 4KB row in
// registers; one pass over HBM (268 MB -> ~11.5 us floor at 23.3 TB/s).
__global__ void __launch_bounds__(256)
focal_rows_kernel(const float* __restrict__ inp, const int* __restrict__ target,
                  const int* __restrict__ counts, const float* __restrict__ minv,
                  float* __restrict__ partials, int B) {
    const int lane = threadIdx.x & 31;
    const int wave = threadIdx.x >> 5;
    const int row  = blockIdx.x * 8 + wave;
    const bool valid = (row < B);
    const float* rp = inp + (size_t)row * C_DIM;

    float4 v[8];
    if (valid) {
        const float4* rp4 = (const float4*)rp;
        #pragma unroll
        for (int i = 0; i < 8; ++i)
            v[i] = rp4[i * 32 + lane];            // 512B per wave per iter, coalesced
    } else {
        #pragma unroll
        for (int i = 0; i < 8; ++i) v[i] = make_float4(0.f, 0.f, 0.f, 0.f);
    }

    // row max
    float mx = -INFINITY;
    #pragma unroll
    for (int i = 0; i < 8; ++i)
        mx = fmaxf(mx, fmaxf(fmaxf(v[i].x, v[i].y), fmaxf(v[i].z, v[i].w)));
    #pragma unroll
    for (int off = 16; off; off >>= 1) mx = fmaxf(mx, __shfl_xor(mx, off, 32));

    // row sum of exp(x - max)
    float s = 0.f;
    #pragma unroll
    for (int i = 0; i < 8; ++i)
        s += __expf(v[i].x - mx) + __expf(v[i].y - mx)
           + __expf(v[i].z - mx) + __expf(v[i].w - mx);
    #pragma unroll
    for (int off = 16; off; off >>= 1) s += __shfl_xor(s, off, 32);

    __shared__ float ls[8];
    if (lane == 0) {
        float loss = 0.f;
        if (valid) {
            int   t     = target[row];
            float xt    = rp[t];                  // 4B reload, resident in 192MB L2
            float logpt = (xt - mx) - __logf(s);
            float pt    = __expf(logpt);
            float cw    = (*minv) / (float)counts[t] + EPSV;
            float om    = 1.f - pt;
            loss = -(om * om) * logpt * cw;       // GAMMA = 2
        }
        ls[wave] = loss;
    }
    __syncthreads();
    if (threadIdx.x == 0) {
        float acc = 0.f;
        #pragma unroll
        for (int i = 0; i < 8; ++i) acc += ls[i];
        partials[blockIdx.x] = acc;
    }
}

// Final reduction of per-block partials. The last 8->1 stage runs as a real
// V_WMMA_F32_16X16X4_F32 (A = partials in the 16x4 f32 layout, B = ones, so
// D[m,n] = sum_k A[m,k]); full f32 precision, EXEC all-1s (whole wave 0 runs it).
__global__ void __launch_bounds__(256)
final_reduce_kernel(const float* __restrict__ partials, int n,
                    float* __restrict__ out, float invB) {
    const int lane = threadIdx.x & 31;
    const int wave = threadIdx.x >> 5;

    float s = 0.f;
    for (int i = threadIdx.x; i < n; i += 256) s += partials[i];
    #pragma unroll
    for (int off = 16; off; off >>= 1) s += __shfl_xor(s, off, 32);

    __shared__ float wsum[16];
    if (threadIdx.x < 16) wsum[threadIdx.x] = 0.f;
    __syncthreads();
    if (lane == 0) wsum[wave] = s;
    __syncthreads();

    if (threadIdx.x < 32) {          // wave 0 only; EXEC == all ones within it
        // A (16x4 f32): VGPR0 lanes0-15 -> A[m=lane,k=0]; lanes16-31 -> A[m,k=2] (zeroed)
        v2f a; a.x = (lane < 16) ? wsum[lane] : 0.f; a.y = 0.f;
        v2f b; b.x = 1.f; b.y = 1.f;                  // B[k,n] = 1 -> D[m,n] = A[m,0]
        v8f c = {};
        c = __builtin_amdgcn_wmma_f32_16x16x4_f32(
                /*neg_a=*/false, a, /*neg_b=*/false, b,
                /*c_mod=*/(short)0, c, /*reuse_a=*/false, /*reuse_b=*/false);
        // D VGPR j: lanes0-15 hold D[j,n]=wsum[j]; lanes16-31 hold D[j+8,n]=wsum[j+8]
        float tot = c[0]+c[1]+c[2]+c[3]+c[4]+c[5]+c[6]+c[7];
        tot += __shfl_xor(tot, 16, 32);               // combine halves (upper half is 0)
        if (threadIdx.x == 0) out[0] = tot * invB;
    }
}

extern "C" void kernel_launch(void* const* d_in, const int* in_sizes, int n_in,
                              void* d_out, int out_size, void* d_ws, size_t ws_size,
                              hipStream_t stream) {
    const float* inp    = (const float*)d_in[0];
    const int*   target = (const int*)d_in[1];
    const int    B      = in_sizes[1];            // 65536

    int*   counts   = (int*)d_ws;                 // 1024 ints
    float* minv     = (float*)d_ws + C_DIM;       // 1 float
    float* partials = (float*)d_ws + C_DIM + 64;  // B/8 floats

    const int nblocks = (B + 7) / 8;

    zero_counts_kernel<<<(C_DIM + 255) / 256, 256, 0, stream>>>(counts);
    hist_kernel<<<(B + 255) / 256, 256, 0, stream>>>(target, counts, B);
    min_kernel<<<1, 256, 0, stream>>>(counts, minv);
    focal_rows_kernel<<<nblocks, 256, 0, stream>>>(inp, target, counts, minv, partials, B);
    final_reduce_kernel<<<1, 256, 0, stream>>>(partials, nblocks, (float*)d_out,
                                               1.0f / (float)B);
}